// KnnRegressFromGED_64304250355827
// MI455X (gfx1250) — compile-verified
//
#include <hip/hip_runtime.h>
#include <stdint.h>

// ---------------------------------------------------------------------------
// KNN regression from GED matrix, MI455X (gfx1250).
//   ged : [50000, 50000 rows x 4096 cols] f32, y : [50000] f32 -> out [4096] f32
// Bandwidth-bound: single fused pass over ged (819 MB ~ 35us @ 23.3 TB/s).
// Per-column normalization is a positive scale -> top-k on raw values, rescale
// the selected values by 1/max(norm,eps) in the merge kernel.
// CDNA5 paths: async global->LDS staging (ASYNCcnt) + WMMA f32 16x16x4 with
// A=ones for the column sum-of-squares on the matrix pipe.
// ---------------------------------------------------------------------------

typedef __attribute__((ext_vector_type(2))) float v2f;
typedef __attribute__((ext_vector_type(8))) float v8f;

#define N_TRAIN 50000
#define N_TEST  4096
#define KSEL    32
#define COLS    128                       // columns per block (1 per thread)
#define BLOCKT  128                       // 4 waves (wave32)
#define NCHUNK  25                        // row chunks (grid.y)
#define ROWS_PER_CHUNK (N_TRAIN / NCHUNK) // 2000
#define STAGE_ROWS 8
#define NSTAGES (ROWS_PER_CHUNK / STAGE_ROWS) // 250
#define LOADS_PER_THREAD ((STAGE_ROWS * COLS) / (4 * BLOCKT)) // 2 x b128

// Async bulk copy: 16B per lane, global -> LDS, tracked by ASYNCcnt.
__device__ __forceinline__ void async_ld16(uint32_t lds_addr, const void* gaddr) {
  asm volatile("global_load_async_to_lds_b128 %0, %1, off"
               :: "v"(lds_addr), "v"((uint64_t)gaddr)
               : "memory");
}

#if __has_builtin(__builtin_amdgcn_s_wait_asynccnt)
#define WAIT_ASYNC(n) __builtin_amdgcn_s_wait_asynccnt(n)
#else
#define WAIT_ASYNC(n) asm volatile("s_wait_asynccnt %0" :: "i"(n) : "memory")
#endif

// ---------------------------------------------------------------------------
// Kernel 1: fused streaming pass.
//  - double-buffered async staging of 8x128 f32 tiles into LDS
//  - per-thread (one column) running sorted top-32 smallest (LDS lists,
//    register guard on the 32nd value so steady-state cost is 1 compare/elem)
//  - column sum-of-squares via V_WMMA_F32_16X16X4_F32 with A = ones:
//    D[m][n] = sum_k B[k][n]; accumulate C over the whole chunk.
// ---------------------------------------------------------------------------
__global__ void __launch_bounds__(BLOCKT)
knn_partial(const float* __restrict__ ged,
            float* __restrict__ pv, int* __restrict__ pidx,
            float* __restrict__ pss)
{
  __shared__ float stage[2][STAGE_ROWS * COLS]; // 8 KB
  __shared__ float lv[BLOCKT * KSEL];           // 16 KB  per-thread top-32 vals
  __shared__ int   li[BLOCKT * KSEL];           // 16 KB  per-thread top-32 idx

  const int tid     = threadIdx.x;
  const int col0    = blockIdx.x * COLS;
  const int rowBase = blockIdx.y * ROWS_PER_CHUNK;
  const int lane    = tid & 31;
  const int wv      = tid >> 5;          // wave id: covers cols [wv*32, wv*32+32)
  const int nIdx    = lane & 15;         // B-matrix column within 16-col chunk
  const int kBase   = (lane >> 4) * 2;   // B-matrix K rows held by this lane

  const float INF = __builtin_inff();
  const int base  = tid * KSEL;
#pragma unroll
  for (int p = 0; p < KSEL; ++p) { lv[base + p] = INF; li[base + p] = 0; }
  float kth = INF;

  v8f acc0 = {};            // sum-of-squares for cols wv*32 + 0..15
  v8f acc1 = {};            // sum-of-squares for cols wv*32 + 16..31
  v2f aone = {1.0f, 1.0f};  // A = ones (layout-independent)

  auto issue = [&](int s) {
    float* sb = stage[s & 1];
    const int r0 = rowBase + s * STAGE_ROWS;
#pragma unroll
    for (int j = 0; j < LOADS_PER_THREAD; ++j) {
      const int c = j * BLOCKT + tid;   // 16B packet id, 32 packets per row
      const int r = c >> 5;
      const int f = (c & 31) * 4;       // float offset within the row
      const float* g = ged + (size_t)(r0 + r) * N_TEST + (col0 + f);
      async_ld16((uint32_t)(uintptr_t)(sb + c * 4), g);
    }
  };

  issue(0);
  for (int s = 0; s < NSTAGES; ++s) {
    if (s + 1 < NSTAGES) { issue(s + 1); WAIT_ASYNC(2); }  // stage s complete (in-order)
    else                 { WAIT_ASYNC(0); }
    __syncthreads();
    const float* sb = stage[s & 1];

    // ---- matrix-pipe Sigma x^2: D = ones(16x4) * (B o B)(4x16) + C ----
#pragma unroll
    for (int r0 = 0; r0 < STAGE_ROWS; r0 += 4) {
      float x0 = sb[(r0 + kBase + 0) * COLS + wv * 32 + nIdx];
      float x1 = sb[(r0 + kBase + 1) * COLS + wv * 32 + nIdx];
      v2f b0 = {x0 * x0, x1 * x1};
      acc0 = __builtin_amdgcn_wmma_f32_16x16x4_f32(false, aone, false, b0,
                                                   (short)0, acc0, false, false);
      float x2 = sb[(r0 + kBase + 0) * COLS + wv * 32 + 16 + nIdx];
      float x3 = sb[(r0 + kBase + 1) * COLS + wv * 32 + 16 + nIdx];
      v2f b1 = {x2 * x2, x3 * x3};
      acc1 = __builtin_amdgcn_wmma_f32_16x16x4_f32(false, aone, false, b1,
                                                   (short)0, acc1, false, false);
    }

    // ---- per-thread running top-32 smallest for its column ----
    const int rowS = rowBase + s * STAGE_ROWS;
#pragma unroll
    for (int r = 0; r < STAGE_ROWS; ++r) {
      const float v = sb[r * COLS + tid];
      if (v < kth) {                       // rare in steady state (~K/t)
        int p = KSEL - 1;
        while (p > 0) {
          const float q = lv[base + p - 1];
          if (q <= v) break;
          lv[base + p] = q; li[base + p] = li[base + p - 1];
          --p;
        }
        lv[base + p] = v; li[base + p] = rowS + r;
        kth = lv[base + KSEL - 1];
      }
    }
    __syncthreads();  // all waves done with this buffer before it is re-issued
  }

  // C/D layout: lane n, VGPR0 = D[M][N=n] and all D rows are identical, so
  // every lane reads its own column's sum with no shuffles.
  const float ss = (lane < 16) ? acc0[0] : acc1[0];
  const int col  = col0 + tid;
  const size_t o = (size_t)blockIdx.y * N_TEST + col;
  pss[o] = ss;
#pragma unroll
  for (int p = 0; p < KSEL; ++p) {
    pv[o * KSEL + p]   = lv[base + p];
    pidx[o * KSEL + p] = li[base + p];
  }
}

// ---------------------------------------------------------------------------
// Kernel 2: per-column merge of 25 partial top-32 lists (800 candidates),
// norm from summed partial Sigma x^2, then weighted average of y[ind].
// One wave (32 lanes) per column; wave-butterfly reductions.
// ---------------------------------------------------------------------------
__global__ void __launch_bounds__(32)
knn_merge(const float* __restrict__ pv, const int* __restrict__ pidx,
          const float* __restrict__ pss, const float* __restrict__ y,
          float* __restrict__ out)
{
  const int col  = blockIdx.x;
  const int lane = threadIdx.x;
  __shared__ float cv[NCHUNK * KSEL];
  __shared__ int   ci[NCHUNK * KSEL];
  __shared__ float selv[KSEL];
  __shared__ int   seli[KSEL];

  for (int j = lane; j < NCHUNK * KSEL; j += 32) {
    const int ch = j / KSEL, p = j - ch * KSEL;
    const size_t o = ((size_t)ch * N_TEST + col) * KSEL + p;
    cv[j] = pv[o];
    ci[j] = pidx[o];
  }
  float ss = (lane < NCHUNK) ? pss[(size_t)lane * N_TEST + col] : 0.0f;
#pragma unroll
  for (int off = 16; off > 0; off >>= 1) ss += __shfl_xor(ss, off, 32);
  __syncthreads();

  const float INF = __builtin_inff();
  for (int k = 0; k < KSEL; ++k) {
    float mv = INF; int mp = 0;
    for (int j = lane; j < NCHUNK * KSEL; j += 32) {
      const float v = cv[j];
      if (v < mv) { mv = v; mp = j; }
    }
#pragma unroll
    for (int off = 16; off > 0; off >>= 1) {
      const float ov = __shfl_xor(mv, off, 32);
      const int   op = __shfl_xor(mp, off, 32);
      if (ov < mv) { mv = ov; mp = op; }
    }
    if (lane == 0) { selv[k] = mv; seli[k] = ci[mp]; cv[mp] = INF; }
    __syncthreads();
  }

  const float dn  = fmaxf(sqrtf(ss), 1e-12f);   // matches jnp.maximum(norm, EPS)
  const float v   = selv[lane] / dn;            // normalized distance
  const float sim = 1.0f / (v + 1.0f);
  float num = sim * y[seli[lane]];
  float den = sim;
#pragma unroll
  for (int off = 16; off > 0; off >>= 1) {
    num += __shfl_xor(num, off, 32);
    den += __shfl_xor(den, off, 32);
  }
  if (lane == 0) out[col] = num / den;
}

// ---------------------------------------------------------------------------
extern "C" void kernel_launch(void* const* d_in, const int* in_sizes, int n_in,
                              void* d_out, int out_size, void* d_ws, size_t ws_size,
                              hipStream_t stream) {
  const float* ged = (const float*)d_in[0];
  const float* y   = (const float*)d_in[1];
  float* out       = (float*)d_out;

  // workspace: [pv: NCHUNK*N_TEST*K f32][pidx: same i32][pss: NCHUNK*N_TEST f32]
  const size_t PVN = (size_t)N_TEST * NCHUNK * KSEL;
  float* pv  = (float*)d_ws;
  int*   pidx = (int*)((char*)d_ws + PVN * sizeof(float));
  float* pss = (float*)((char*)d_ws + 2 * PVN * sizeof(float));
  (void)in_sizes; (void)n_in; (void)out_size; (void)ws_size;

  dim3 g1(N_TEST / COLS, NCHUNK);                 // 32 x 25 = 800 blocks
  knn_partial<<<g1, BLOCKT, 0, stream>>>(ged, pv, pidx, pss);
  knn_merge<<<N_TEST, 32, 0, stream>>>(pv, pidx, pss, y, out);
}